// aspect_rs_61538291417368
// MI455X (gfx1250) — compile-verified
//
#include <hip/hip_runtime.h>
#include <hip/hip_bf16.h>

// Problem sizes (from reference)
#define RR 20000   // reviews
#define SS 64      // words per review
#define DD 128     // embedding dim
#define KK 32      // aspects
#define BB 2048    // batch
#define LL 20480   // history length

typedef float v2f __attribute__((ext_vector_type(2)));
typedef float v8f __attribute__((ext_vector_type(8)));

static __device__ __forceinline__ v8f v8_splat(float x) {
  v8f r = {x, x, x, x, x, x, x, x};
  return r;
}

// D = A(16x4) * B(4x16) + C, fp32 WMMA (CDNA5, wave32)
static __device__ __forceinline__ v8f wmma_f32_4(v2f a, v2f b, v8f c) {
  return __builtin_amdgcn_wmma_f32_16x16x4_f32(false, a, false, b, (short)0, c,
                                               false, false);
}

// ---------------------------------------------------------------------------
// Kernel 1: per-16-review tile, one wave32 per workgroup.
//  V = M_w @ Y            (WMMA, 128x16x128)
//  dx = E . v ; softmax_s ; z = sum_s ax*E   (streaming, E cached in LDS)
//  p  = softmax_K(Z @ W_w + W_b)             (WMMA, 16x32x128)
//  rs = p @ T_w + T_b                        (WMMA, 16x128x32)
// Note: the M_b . y_s term is constant over s -> cancels in softmax, skipped.
// ---------------------------------------------------------------------------
__global__ __launch_bounds__(32) void aspect_reviews_kernel(
    const int* __restrict__ hist_rev,    // [R,S]
    const float* __restrict__ y_s,       // [R,D]
    const float* __restrict__ word_emb,  // [V,D]
    const float* __restrict__ M_w,       // [D,D]
    const float* __restrict__ W_w,       // [D,K]
    const float* __restrict__ W_b,       // [K]
    const float* __restrict__ T_w,       // [K,D]
    const float* __restrict__ T_b,       // [D]
    float* __restrict__ rs)              // [R,D] out
{
  __shared__ float bufA[16][132];  // Y tile, later reused as Z tile (padded)
  __shared__ float Vlds[16][132];  // V = M_w @ Y, stored [review][d]
  __shared__ float Elds[64][132];  // gathered word embeddings of one review
  __shared__ float AXl[16][64];    // word-attention weights
  __shared__ float Plds[16][36];   // aspect probabilities (padded)

  const int lane = threadIdx.x;    // 0..31
  const int half = lane >> 4;      // 0/1
  const int l16  = lane & 15;
  const int r0   = blockIdx.x * 16;

  // ---- load Y tile: y_s[r0..r0+15][0..127] ----
  for (int i = lane; i < 16 * 32; i += 32) {
    const int row = i >> 5, c4 = i & 31;
    const float4 v = *(const float4*)(y_s + (size_t)(r0 + row) * DD + c4 * 4);
    *(float4*)(&bufA[row][c4 * 4]) = v;
  }
  __syncthreads();

  // ---- V = M_w @ Y  (output 128x16; 8 M-tiles x 32 K-steps of WMMA) ----
  for (int mt = 0; mt < 8; ++mt) {
    v8f acc = v8_splat(0.0f);
    for (int kt = 0; kt < 32; ++kt) {
      // A[m][k] = M_w[mt*16+m][kt*4+k] ; lane holds M=l16, K={2h,2h+1}
      const float* ap = M_w + (size_t)(mt * 16 + l16) * DD + kt * 4 + half * 2;
      v2f a; a.x = ap[0]; a.y = ap[1];
      // B[k][n] = Y[kt*4+k][n]  (n = review l16)
      v2f b; b.x = bufA[l16][kt * 4 + half * 2];
             b.y = bufA[l16][kt * 4 + half * 2 + 1];
      acc = wmma_f32_4(a, b, acc);
    }
    // C/D layout: VGPR j -> row M = j + 8*half, col N = l16 (review)
#pragma unroll
    for (int j = 0; j < 8; ++j)
      Vlds[l16][mt * 16 + half * 8 + j] = acc[j];
  }
  __syncthreads();

  // ---- per-review: dx, softmax_s, z ----
  for (int n = 0; n < 16; ++n) {
    float dxv[2];
#pragma unroll
    for (int t = 0; t < 2; ++t) {
      const int s = lane + t * 32;
      const int w = hist_rev[(size_t)(r0 + n) * SS + s];
      const float4* wp = (const float4*)(word_emb + (size_t)w * DD);
      float acc = 0.0f;
#pragma unroll 8
      for (int q = 0; q < 32; ++q) {
        const float4 e = wp[q];
        *(float4*)(&Elds[s][q * 4]) = e;  // cache for z-pass (avoid 2nd gather)
        const float4 v = *(const float4*)(&Vlds[n][q * 4]);  // LDS broadcast
        acc += e.x * v.x + e.y * v.y + e.z * v.z + e.w * v.w;
      }
      dxv[t] = acc;
    }
    // softmax over the 64 s-values spread across the wave (2 per lane)
    float mx = fmaxf(dxv[0], dxv[1]);
    for (int m = 16; m >= 1; m >>= 1) mx = fmaxf(mx, __shfl_xor(mx, m, 32));
    const float e0 = __expf(dxv[0] - mx), e1 = __expf(dxv[1] - mx);
    float sm = e0 + e1;
    for (int m = 16; m >= 1; m >>= 1) sm += __shfl_xor(sm, m, 32);
    const float inv = 1.0f / sm;
    AXl[n][lane]      = e0 * inv;
    AXl[n][lane + 32] = e1 * inv;
    __syncthreads();

    // z[d] = sum_s ax[s] * E[s][d] ; lane owns d = lane*4 .. lane*4+3
    float4 z = {0.f, 0.f, 0.f, 0.f};
#pragma unroll 8
    for (int s = 0; s < 64; ++s) {
      const float a = AXl[n][s];  // broadcast
      const float4 e = *(const float4*)(&Elds[s][lane * 4]);
      z.x += a * e.x; z.y += a * e.y; z.z += a * e.z; z.w += a * e.w;
    }
    *(float4*)(&bufA[n][lane * 4]) = z;  // Z tile overwrites Y tile
    __syncthreads();
  }

  // ---- logits = Z @ W_w + W_b ; softmax over K=32 ----
  v8f lg[2];
  for (int nt = 0; nt < 2; ++nt) {
    v8f acc = v8_splat(W_b[nt * 16 + l16]);  // bias depends on column only
    for (int kt = 0; kt < 32; ++kt) {
      v2f a; a.x = bufA[l16][kt * 4 + half * 2];
             a.y = bufA[l16][kt * 4 + half * 2 + 1];
      const float* bp = W_w + (size_t)(kt * 4 + half * 2) * KK + nt * 16 + l16;
      v2f b; b.x = bp[0]; b.y = bp[KK];
      acc = wmma_f32_4(a, b, acc);
    }
    lg[nt] = acc;
  }
  // softmax per review row (= j + 8*half); 32 aspects across 16 lanes x 2 tiles
#pragma unroll
  for (int j = 0; j < 8; ++j) {
    float mx = fmaxf(lg[0][j], lg[1][j]);
    for (int m = 8; m >= 1; m >>= 1) mx = fmaxf(mx, __shfl_xor(mx, m, 32));
    const float e0 = __expf(lg[0][j] - mx), e1 = __expf(lg[1][j] - mx);
    float sm = e0 + e1;
    for (int m = 8; m >= 1; m >>= 1) sm += __shfl_xor(sm, m, 32);
    const float inv = 1.0f / sm;
    Plds[j + 8 * half][l16]      = e0 * inv;
    Plds[j + 8 * half][16 + l16] = e1 * inv;
  }
  __syncthreads();

  // ---- rs = P @ T_w + T_b  (16x128, K=32) ----
  for (int nt = 0; nt < 8; ++nt) {
    v8f acc = v8_splat(T_b[nt * 16 + l16]);
    for (int kt = 0; kt < 8; ++kt) {
      v2f a; a.x = Plds[l16][kt * 4 + half * 2];
             a.y = Plds[l16][kt * 4 + half * 2 + 1];
      const float* bp = T_w + (size_t)(kt * 4 + half * 2) * DD + nt * 16 + l16;
      v2f b; b.x = bp[0]; b.y = bp[DD];
      acc = wmma_f32_4(a, b, acc);
    }
#pragma unroll
    for (int j = 0; j < 8; ++j)
      rs[(size_t)(r0 + j + 8 * half) * DD + nt * 16 + l16] = acc[j];
  }
}

// ---------------------------------------------------------------------------
// Kernel 2: segment-sum of gathered rs rows (user + item) via float atomics.
// blocks [0,L) -> user entries; [L,2L) -> item entries. 128 threads = one d each.
// ---------------------------------------------------------------------------
__global__ __launch_bounds__(128) void seg_acc_kernel(
    const float* __restrict__ rs,
    const int* __restrict__ uidx, const int* __restrict__ useg,
    const int* __restrict__ iidx, const int* __restrict__ iseg,
    float* __restrict__ uacc, float* __restrict__ iacc,
    float* __restrict__ ucnt, float* __restrict__ icnt)
{
  const int l = blockIdx.x;
  const int d = threadIdx.x;
  if (l < LL) {
    const int idx = uidx[l], seg = useg[l];
    atomicAdd(&uacc[(size_t)seg * DD + d], rs[(size_t)idx * DD + d]);
    if (d == 0) atomicAdd(&ucnt[seg], 1.0f);
  } else {
    const int l2 = l - LL;
    const int idx = iidx[l2], seg = iseg[l2];
    atomicAdd(&iacc[(size_t)seg * DD + d], rs[(size_t)idx * DD + d]);
    if (d == 0) atomicAdd(&icnt[seg], 1.0f);
  }
}

// ---------------------------------------------------------------------------
// Kernel 3: prediction per batch element (one wave32 per b).
// pred[b] = (ua/cu).(ia/ci) + uf.itf + avg
// ---------------------------------------------------------------------------
__global__ __launch_bounds__(32) void predict_kernel(
    const int* __restrict__ user, const int* __restrict__ item,
    const float* __restrict__ uacc, const float* __restrict__ iacc,
    const float* __restrict__ ucnt, const float* __restrict__ icnt,
    const float* __restrict__ user_emb, const float* __restrict__ item_emb,
    const float* __restrict__ avg, float* __restrict__ out)
{
  const int b = blockIdx.x;
  const int lane = threadIdx.x;
  const float cu = fmaxf(ucnt[b], 1.0f), ci = fmaxf(icnt[b], 1.0f);
  const float inv = 1.0f / (cu * ci);
  const int u = user[b], it = item[b];
  float s = 0.0f;
#pragma unroll
  for (int t = 0; t < 4; ++t) {
    const int d = lane + t * 32;
    s += uacc[(size_t)b * DD + d] * iacc[(size_t)b * DD + d] * inv;
    s += user_emb[(size_t)u * DD + d] * item_emb[(size_t)it * DD + d];
  }
  for (int m = 16; m >= 1; m >>= 1) s += __shfl_xor(s, m, 32);
  if (lane == 0) out[b] = s + avg[0];
}

// ---------------------------------------------------------------------------
extern "C" void kernel_launch(void* const* d_in, const int* in_sizes, int n_in,
                              void* d_out, int out_size, void* d_ws, size_t ws_size,
                              hipStream_t stream) {
  (void)in_sizes; (void)n_in; (void)out_size; (void)ws_size;
  const int*   user      = (const int*)d_in[0];
  const int*   item      = (const int*)d_in[1];
  const int*   hist_rev  = (const int*)d_in[2];
  const float* y_s       = (const float*)d_in[3];
  const int*   uidx      = (const int*)d_in[4];
  const int*   useg      = (const int*)d_in[5];
  const int*   iidx      = (const int*)d_in[6];
  const int*   iseg      = (const int*)d_in[7];
  const float* word_emb  = (const float*)d_in[8];
  const float* M_w       = (const float*)d_in[9];
  /* d_in[10] = M_b: constant-per-review term cancels in word softmax */
  const float* W_w       = (const float*)d_in[11];
  const float* W_b       = (const float*)d_in[12];
  const float* T_w       = (const float*)d_in[13];
  const float* T_b       = (const float*)d_in[14];
  const float* user_emb  = (const float*)d_in[15];
  const float* item_emb  = (const float*)d_in[16];
  const float* avg       = (const float*)d_in[17];
  float* out = (float*)d_out;

  // Workspace layout (≈12.3 MB): rs | uacc | iacc | ucnt | icnt
  float* rs   = (float*)d_ws;
  float* uacc = rs + (size_t)RR * DD;
  float* iacc = uacc + (size_t)BB * DD;
  float* ucnt = iacc + (size_t)BB * DD;
  float* icnt = ucnt + BB;

  // zero accumulators every launch (graph-capturable async memset)
  hipMemsetAsync(uacc, 0, ((size_t)2 * BB * DD + 2 * BB) * sizeof(float), stream);

  aspect_reviews_kernel<<<RR / 16, 32, 0, stream>>>(
      hist_rev, y_s, word_emb, M_w, W_w, W_b, T_w, T_b, rs);
  seg_acc_kernel<<<2 * LL, 128, 0, stream>>>(
      rs, uidx, useg, iidx, iseg, uacc, iacc, ucnt, icnt);
  predict_kernel<<<BB, 32, 0, stream>>>(
      user, item, uacc, iacc, ucnt, icnt, user_emb, item_emb, avg, out);
}